// PairwiseAtomicEmbedding_5832565588253
// MI455X (gfx1250) — compile-verified
//
#include <hip/hip_runtime.h>
#include <math.h>

#define NRES   8192
#define NEDGE  262144
#define FEAT   596
#define FEPS   1e-8f

typedef float v2f __attribute__((ext_vector_type(2)));
typedef float v8f __attribute__((ext_vector_type(8)));

// fast HW sqrt: single v_sqrt_f32 (inputs here are O(1..1e3), no denorms)
__device__ __forceinline__ float fsqrt_hw(float x) {
    return __builtin_amdgcn_sqrtf(x);
}

// ---------------- workspace layout (floats) ----------------
// resR     : NRES*9   rotation matrices (row-major R[j][i] stored R[3*j+i])
// resT     : NRES*3   translations
// resBB    : NRES*12  backbone N,CA,C,CB(virtual) xyz
// resLocal : NRES*42  atom14 in local frame
// resUN    : NRES*14  |local + eps|^2 per atom
#define WS_R   0
#define WS_T   (NRES*9)
#define WS_BB  (NRES*12)
#define WS_LOC (NRES*24)
#define WS_UN  (NRES*66)

__global__ __launch_bounds__(256)
void residue_precompute(const float* __restrict__ atom14,
                        const float* __restrict__ rigids7,
                        float* __restrict__ ws) {
    int n = blockIdx.x * blockDim.x + threadIdx.x;
    if (n >= NRES) return;

    const float* q = rigids7 + (size_t)n * 7;
    float r = q[0], i = q[1], j = q[2], k = q[3];
    float tx = q[4], ty = q[5], tz = q[6];

    // quaternion -> rotation (row-major)
    float R00 = 1.f - 2.f*(j*j + k*k);
    float R01 = 2.f*(i*j - k*r);
    float R02 = 2.f*(i*k + j*r);
    float R10 = 2.f*(i*j + k*r);
    float R11 = 1.f - 2.f*(i*i + k*k);
    float R12 = 2.f*(j*k - i*r);
    float R20 = 2.f*(i*k - j*r);
    float R21 = 2.f*(j*k + i*r);
    float R22 = 1.f - 2.f*(i*i + j*j);

    float* resR  = ws + WS_R   + (size_t)n * 9;
    float* resT  = ws + WS_T   + (size_t)n * 3;
    float* resBB = ws + WS_BB  + (size_t)n * 12;
    float* resL  = ws + WS_LOC + (size_t)n * 42;
    float* resU  = ws + WS_UN  + (size_t)n * 14;

    resR[0]=R00; resR[1]=R01; resR[2]=R02;
    resR[3]=R10; resR[4]=R11; resR[5]=R12;
    resR[6]=R20; resR[7]=R21; resR[8]=R22;
    resT[0]=tx; resT[1]=ty; resT[2]=tz;

    const float* A = atom14 + (size_t)n * 42;
    #pragma unroll 2
    for (int a = 0; a < 14; ++a) {
        float x = A[a*3+0] - tx, y = A[a*3+1] - ty, z = A[a*3+2] - tz;
        // local_i = sum_j R[j][i] * x_j  (R^T x)
        float lx = R00*x + R10*y + R20*z;
        float ly = R01*x + R11*y + R21*z;
        float lz = R02*x + R12*y + R22*z;
        resL[a*3+0] = lx; resL[a*3+1] = ly; resL[a*3+2] = lz;
        float ux = lx + FEPS, uy = ly + FEPS, uz = lz + FEPS;
        resU[a] = ux*ux + uy*uy + uz*uz;
    }

    // backbone + ideal virtual CB
    float nx = A[0], ny = A[1], nz = A[2];
    float cax = A[3], cay = A[4], caz = A[5];
    float cx = A[6], cy = A[7], cz = A[8];
    float bx = cax - nx, by = cay - ny, bz = caz - nz;
    float cvx = cx - cax, cvy = cy - cay, cvz = cz - caz;
    float ax = by*cvz - bz*cvy;
    float ay = bz*cvx - bx*cvz;
    float az = bx*cvy - by*cvx;
    float cbx = -0.58273431f*ax + 0.56802827f*bx - 0.54067466f*cvx + cax;
    float cby = -0.58273431f*ay + 0.56802827f*by - 0.54067466f*cvy + cay;
    float cbz = -0.58273431f*az + 0.56802827f*bz - 0.54067466f*cvz + caz;
    resBB[0]=nx;  resBB[1]=ny;  resBB[2]=nz;
    resBB[3]=cax; resBB[4]=cay; resBB[5]=caz;
    resBB[6]=cx;  resBB[7]=cy;  resBB[8]=cz;
    resBB[9]=cbx; resBB[10]=cby; resBB[11]=cbz;
}

__global__ __launch_bounds__(256)
void edge_features(const int* __restrict__ ei,
                   const int* __restrict__ seq,
                   const int* __restrict__ nmask,
                   const float* __restrict__ atom14,
                   const float* __restrict__ ws,
                   float* __restrict__ out) {
    __shared__ __align__(16) float srow[8][FEAT];   // staged output rows
    __shared__ float su[8][16][3];                  // src local + eps (padded)
    __shared__ float sun[8][16];                    // |u'|^2
    __shared__ float sv[8][16][3];                  // dst atoms in src frame
    __shared__ float svn[8][16];                    // |v|^2
    __shared__ float sdist[8][16];                  // 4x4 backbone distances

    const float* resR  = ws + WS_R;
    const float* resT  = ws + WS_T;
    const float* resBB = ws + WS_BB;
    const float* resL  = ws + WS_LOC;
    const float* resU  = ws + WS_UN;

    int lane = threadIdx.x & 31;
    int w    = threadIdx.x >> 5;
    int e    = blockIdx.x * 8 + w;

    int dst = ei[e];
    int src = ei[NEDGE + e];
    int nm_s = nmask[src];
    int nm_d = nmask[dst];
    float msrc = nm_s ? 0.f : 1.f;
    float mdst = nm_d ? 0.f : 1.f;

    // reference indexes unnoised_edges[-1]: one global scalar (L2-hot loads)
    int dlast = ei[NEDGE - 1];
    int slast = ei[2 * NEDGE - 1];
    float pdScale = (nmask[dlast] || nmask[slast]) ? 0.f : 1.f;

    float* row = srow[w];

    // ---- phase A: stage scalars, one-hots, pos-embed, locals ----
    if (lane == 0) { row[0] = nm_d ? 1.f : 0.f; row[1] = nm_s ? 1.f : 0.f; }

    int sseq = seq[src], dseq = seq[dst];
    if (lane < 21) {
        row[2  + lane] = (lane == sseq) ? msrc : 0.f;
        row[23 + lane] = (lane == dseq) ? mdst : 0.f;
    }

    if (lane < 16) {
        // positional embedding: freq_k = 10000^{-k/8}
        int kk = lane & 7;
        float d = (float)(dst - src);
        float freq = __expf(-(float)kk * (9.210340371976184f / 8.f));
        float ang = d * freq;
        row[300 + lane] = (lane < 8) ? __cosf(ang) : __sinf(ang);

        // 4x4 backbone distances (pair = a*4+b)
        int a = lane >> 2, b = lane & 3;
        const float* bs = resBB + (size_t)src * 12 + a * 3;
        const float* bd = resBB + (size_t)dst * 12 + b * 3;
        float dx = bs[0] - bd[0] + FEPS;
        float dy = bs[1] - bd[1] + FEPS;
        float dz = bs[2] - bd[2] + FEPS;
        sdist[w][lane] = fsqrt_hw(dx*dx + dy*dy + dz*dz);

        if (lane < 14) {
            // src atoms in local frame (feat block 316..357) + WMMA A staging
            float lx = resL[(size_t)src*42 + lane*3 + 0];
            float ly = resL[(size_t)src*42 + lane*3 + 1];
            float lz = resL[(size_t)src*42 + lane*3 + 2];
            row[316 + lane*3 + 0] = lx * msrc;
            row[316 + lane*3 + 1] = ly * msrc;
            row[316 + lane*3 + 2] = lz * msrc;
            su[w][lane][0] = lx + FEPS;
            su[w][lane][1] = ly + FEPS;
            su[w][lane][2] = lz + FEPS;
            sun[w][lane]   = resU[(size_t)src*14 + lane];

            // dst atoms transformed into src local frame (358..399) + B staging
            const float* Rp = resR + (size_t)src * 9;
            const float* tp = resT + (size_t)src * 3;
            const float* ap = atom14 + (size_t)dst * 42 + lane * 3;
            float x = ap[0] - tp[0], y = ap[1] - tp[1], z = ap[2] - tp[2];
            float vx = Rp[0]*x + Rp[3]*y + Rp[6]*z;
            float vy = Rp[1]*x + Rp[4]*y + Rp[7]*z;
            float vz = Rp[2]*x + Rp[5]*y + Rp[8]*z;
            row[358 + lane*3 + 0] = vx * mdst;
            row[358 + lane*3 + 1] = vy * mdst;
            row[358 + lane*3 + 2] = vz * mdst;
            sv[w][lane][0] = vx; sv[w][lane][1] = vy; sv[w][lane][2] = vz;
            svn[w][lane]   = vx*vx + vy*vy + vz*vz;
        } else {
            su[w][lane][0] = su[w][lane][1] = su[w][lane][2] = 0.f;
            sv[w][lane][0] = sv[w][lane][1] = sv[w][lane][2] = 0.f;
            sun[w][lane] = 0.f;
            svn[w][lane] = 0.f;
        }
    }
    __syncthreads();

    // ---- phase B: RBF exps + WMMA pairwise distances ----
    #pragma unroll
    for (int it = 0; it < 8; ++it) {
        int o = lane + 32 * it;             // 0..255
        int pair = o >> 4;
        int m = o & 15;
        float dd = sdist[w][pair];
        float mu = 2.f + (20.f / 15.f) * (float)m;
        float td = (dd - mu) * 0.8f;        // /sigma, sigma = 20/16
        row[44 + o] = __expf(-td * td);
    }

    // WMMA: D = A(16x4) * B(4x16) + C, D_ij = |u'_i|^2 - 2 u'_i.v_j + |v_j|^2
    int half = lane >> 4;       // 0: K=0,1 / rows 0..7 ; 1: K=2,3 / rows 8..15
    int ln   = lane & 15;
    v2f Aa, Bb;
    if (half == 0) {
        Aa.x = su[w][ln][0];            // A[M=ln][K=0]
        Aa.y = su[w][ln][1];            // A[M=ln][K=1]
        Bb.x = -2.f * sv[w][ln][0];     // B[K=0][N=ln]
        Bb.y = -2.f * sv[w][ln][1];     // B[K=1][N=ln]
    } else {
        Aa.x = su[w][ln][2];            // A[M=ln][K=2]
        Aa.y = 1.f;                     // A[M=ln][K=3] = 1 (bias trick)
        Bb.x = -2.f * sv[w][ln][2];     // B[K=2][N=ln]
        Bb.y = svn[w][ln];              // B[K=3][N=ln] = |v_N|^2
    }
    v8f Cc;
    #pragma unroll
    for (int r = 0; r < 8; ++r) Cc[r] = sun[w][r + 8 * half];   // |u'_M|^2

    v8f Dd = __builtin_amdgcn_wmma_f32_16x16x4_f32(
        false, Aa, false, Bb, (short)0, Cc, false, false);

    #pragma unroll
    for (int r = 0; r < 8; ++r) {
        int ii = r + 8 * half;
        if (ii < 14 && ln < 14) {
            float d2 = fmaxf(Dd[r], 0.f);
            row[400 + ii * 14 + ln] = fsqrt_hw(d2) * pdScale;
        }
    }
    __syncthreads();

    // ---- phase C: coalesced 128-bit copy of the staged row ----
    float4* orow = (float4*)(out + (size_t)e * FEAT);
    const float4* lrow = (const float4*)row;
    for (int t = lane; t < FEAT / 4; t += 32)   // 149 float4 per edge
        orow[t] = lrow[t];
}

extern "C" void kernel_launch(void* const* d_in, const int* in_sizes, int n_in,
                              void* d_out, int out_size, void* d_ws, size_t ws_size,
                              hipStream_t stream) {
    const int*   edge_index = (const int*)d_in[0];   // (2, NEDGE)
    const int*   seq        = (const int*)d_in[1];   // (NRES,)
    const int*   nmask      = (const int*)d_in[2];   // (NRES,) bool->int
    const float* atom14     = (const float*)d_in[3]; // (NRES,14,3)
    const float* rigids7    = (const float*)d_in[4]; // (NRES,7)
    float* out = (float*)d_out;
    float* ws  = (float*)d_ws;

    residue_precompute<<<NRES / 256, 256, 0, stream>>>(atom14, rigids7, ws);
    edge_features<<<NEDGE / 8, 256, 0, stream>>>(edge_index, seq, nmask,
                                                 atom14, ws, out);
}